// SpatialNorm_68135361184282
// MI455X (gfx1250) — compile-verified
//
#include <hip/hip_runtime.h>
#include <math.h>

// ---------------------------------------------------------------------------
// Problem constants (from reference): B=32, C=256, H=W=64, G=64
//   m = (C/G)*H*W = 16384,  groups-of-4 channels, HW = 4096
// ---------------------------------------------------------------------------
#define BN   32
#define CN   256
#define HWN  4096
#define GN   64
#define MN   16384
#define EPSV 1e-5f

typedef __attribute__((ext_vector_type(2))) float v2f;
typedef __attribute__((ext_vector_type(8))) float v8f;

// ---------------------------------------------------------------------------
// CDNA5 async global->LDS copy (writes LDS straight from the memory pipe,
// tracked on ASYNCcnt; no VGPR staging, no ds_store).
// LDS address operand is the workgroup-relative byte offset = low 32 bits of
// the generic pointer (flat LDS addresses are wg-relative, ISA 10.2).
// ---------------------------------------------------------------------------
__device__ __forceinline__ unsigned lds_off(const void* p) {
  return (unsigned)(unsigned long long)(size_t)p;
}
__device__ __forceinline__ void async_load_b128(unsigned lds_addr, const void* gaddr) {
  asm volatile("global_load_async_to_lds_b128 %0, %1, off"
               :: "v"(lds_addr), "v"(gaddr)
               : "memory");
}
__device__ __forceinline__ void wait_async0() {
  asm volatile("s_wait_asynccnt 0x0" ::: "memory");
}

// ---------------------------------------------------------------------------
// 64x64 fp32 matmul on LDS (or global dest) using V_WMMA_F32_16X16X4_F32.
// 256 threads = 8 waves; each wave computes 2 of the 16 16x16 tiles.
// A,B are 64x64 row-major. EXEC is all-ones at every wmma.
// ---------------------------------------------------------------------------
template <typename PT>
__device__ __forceinline__ void mm64_wmma(PT C, const float* A, const float* B,
                                          int tid) {
  const int lane = tid & 31;
  const int wave = tid >> 5;
  const int l15  = lane & 15;
  const int koff = (lane < 16) ? 0 : 2;
  for (int t = wave; t < 16; t += 8) {
    const int g0 = (t >> 2) << 4;   // row-tile
    const int n0 = (t & 3) << 4;    // col-tile
    v8f acc = {};
    const float* arow = A + (g0 + l15) * GN;
#pragma unroll 4
    for (int kk = 0; kk < GN; kk += 4) {
      v2f av, bv;
      av.x = arow[kk + koff];
      av.y = arow[kk + koff + 1];
      bv.x = B[(kk + koff) * GN + n0 + l15];
      bv.y = B[(kk + koff + 1) * GN + n0 + l15];
      acc = __builtin_amdgcn_wmma_f32_16x16x4_f32(false, av, false, bv,
                                                  (short)0, acc, false, false);
    }
    const int mb = g0 + ((lane < 16) ? 0 : 8);
    const int n  = n0 + l15;
#pragma unroll
    for (int r = 0; r < 8; ++r) C[(mb + r) * GN + n] = acc[r];
  }
}

// ---------------------------------------------------------------------------
// Kernel 0: zero the atomic accumulators (gram + rowsum)
// ---------------------------------------------------------------------------
__global__ __launch_bounds__(256) void k_zero(float* p, int n) {
  int i = blockIdx.x * 256 + threadIdx.x;
  if (i < n) p[i] = 0.0f;
}

// ---------------------------------------------------------------------------
// Kernel 1: per-batch Gram matrix  S[b] += xs_chunk * xs_chunk^T  (WMMA)
//           and row sums (for mean).  Grid: (64 chunks, 32 batches).
//   chunk -> cg = chunk>>4 (channel-in-group), sbase = (chunk&15)*256
// ---------------------------------------------------------------------------
__global__ __launch_bounds__(256) void k_gram(const float* __restrict__ x,
                                              const int* __restrict__ idx,
                                              float* __restrict__ gram,
                                              float* __restrict__ rowsum) {
  __shared__ __align__(16) float sx[GN * 256];   // 64 KB: xs chunk [row g][col c]
  const int chunk = blockIdx.x;
  const int b     = blockIdx.y;
  const int cg    = chunk >> 4;
  const int sbase = (chunk & 15) * 256;
  const int tid   = threadIdx.x;

  // async-stage 64 rows x 256 cols: 4096 x b128, 16 per thread
  const unsigned sxb = lds_off(sx);
  for (int e = tid; e < GN * 64; e += 256) {
    const int h  = e >> 6;            // row
    const int c4 = e & 63;            // 16B chunk within row
    const int ch = idx[4 * h + cg];
    async_load_b128(sxb + (unsigned)e * 16u,
                    x + ((size_t)(b * CN + ch)) * HWN + sbase + c4 * 4);
  }
  wait_async0();
  __syncthreads();

  const int lane = tid & 31;
  const int wave = tid >> 5;
  const int l15  = lane & 15;
  const int koff = (lane < 16) ? 0 : 2;

  for (int t = wave; t < 16; t += 8) {
    const int g0 = (t >> 2) << 4;
    const int h0 = (t & 3) << 4;
    v8f acc = {};
    const float* arow = sx + (g0 + l15) * 256;
    const float* brow = sx + (h0 + l15) * 256;   // B = xs^T -> transposed read
#pragma unroll 4
    for (int kk = 0; kk < 256; kk += 4) {
      v2f av, bv;
      av.x = arow[kk + koff];
      av.y = arow[kk + koff + 1];
      bv.x = brow[kk + koff];
      bv.y = brow[kk + koff + 1];
      acc = __builtin_amdgcn_wmma_f32_16x16x4_f32(false, av, false, bv,
                                                  (short)0, acc, false, false);
    }
    const int mb = g0 + ((lane < 16) ? 0 : 8);
    const int n  = h0 + l15;
#pragma unroll
    for (int r = 0; r < 8; ++r)
      atomicAdd(&gram[(size_t)b * GN * GN + (mb + r) * GN + n], acc[r]);
  }

  // row sums: thread = (row = tid&63, quarter = tid>>6) sums 64 cols
  {
    const int row = tid & 63, q = tid >> 6;
    float s = 0.f;
    const float* p = sx + row * 256 + q * 64;
#pragma unroll 8
    for (int c = 0; c < 64; ++c) s += p[c];
    atomicAdd(&rowsum[b * GN + row], s);
  }
}

// ---------------------------------------------------------------------------
// Kernel 2: finalize sigma, trace-normalize, T=5 Newton-Schulz (WMMA),
//           wm = P * sqrt(trace_inv).  One WG (256 thr) per batch.
// ---------------------------------------------------------------------------
__global__ __launch_bounds__(256) void k_newton(const float* __restrict__ gram,
                                                const float* __restrict__ rowsum,
                                                float* __restrict__ wm,
                                                float* __restrict__ meanout) {
  __shared__ float sN[GN * GN];   // sigma_N
  __shared__ float sP[GN * GN];   // P
  __shared__ float sT1[GN * GN];
  __shared__ float sT2[GN * GN];
  const int b = blockIdx.x, tid = threadIdx.x;
  const float inv_m = 1.0f / (float)MN;

  if (tid < GN) meanout[b * GN + tid] = rowsum[b * GN + tid] * inv_m;

  // sigma (unscaled) = gram/m - mean_i*mean_j ; 16 elements per thread
  const int e0 = tid * 16;
#pragma unroll
  for (int k = 0; k < 16; ++k) {
    const int e = e0 + k, i = e >> 6, j = e & 63;
    const float mi = rowsum[b * GN + i] * inv_m;
    const float mj = rowsum[b * GN + j] * inv_m;
    sN[e] = gram[(size_t)b * GN * GN + e] * inv_m - mi * mj;
  }
  __syncthreads();
  if (tid < GN) sT1[tid] = sN[tid * GN + tid];     // diag
  __syncthreads();
  if (tid == 0) {
    float tr = 0.f;
    for (int i = 0; i < GN; ++i) tr += sT1[i];
    sT1[0] = 1.0f / tr;
  }
  __syncthreads();
  const float tinv = sT1[0];
  __syncthreads();
#pragma unroll
  for (int k = 0; k < 16; ++k) {
    const int e = e0 + k, i = e >> 6, j = e & 63;
    sN[e] *= tinv;
    sP[e] = (i == j) ? 1.0f : 0.0f;
  }
  __syncthreads();

  for (int it = 0; it < 5; ++it) {
    mm64_wmma(sT1, sP, sP, tid);   __syncthreads();   // T1 = P^2
    mm64_wmma(sT2, sT1, sP, tid);  __syncthreads();   // T2 = P^3
    mm64_wmma(sT1, sT2, sN, tid);  __syncthreads();   // T1 = P^3 * sigma_N
#pragma unroll
    for (int k = 0; k < 16; ++k) {
      const int e = e0 + k;
      sP[e] = 1.5f * sP[e] - 0.5f * sT1[e];
    }
    __syncthreads();
  }

  const float ws = sqrtf(tinv);
#pragma unroll
  for (int k = 0; k < 16; ++k) {
    const int e = e0 + k;
    wm[(size_t)b * GN * GN + e] = sP[e] * ws;
  }
}

// ---------------------------------------------------------------------------
// Kernel 3: cross-batch stats: sqrt(var+eps) over B of wm (4096) and mean (64)
// ---------------------------------------------------------------------------
__global__ __launch_bounds__(256) void k_stats(const float* __restrict__ wm,
                                               const float* __restrict__ meanv,
                                               float* __restrict__ svw,
                                               float* __restrict__ svm) {
  const int i = blockIdx.x * 256 + threadIdx.x;
  if (i < GN * GN) {
    float s = 0.f, s2 = 0.f;
    for (int b = 0; b < BN; ++b) {
      const float v = wm[(size_t)b * GN * GN + i];
      s += v; s2 += v * v;
    }
    const float var = (s2 - s * s * (1.0f / BN)) * (1.0f / (BN - 1));
    svw[i] = sqrtf(var + EPSV);
  } else if (i < GN * GN + GN) {
    const int g = i - GN * GN;
    float s = 0.f, s2 = 0.f;
    for (int b = 0; b < BN; ++b) {
      const float v = meanv[b * GN + g];
      s += v; s2 += v * v;
    }
    const float var = (s2 - s * s * (1.0f / BN)) * (1.0f / (BN - 1));
    svm[g] = sqrtf(var + EPSV);
  }
}

// ---------------------------------------------------------------------------
// Kernel 4: gamma = triu(wir)+triu(wir,1)^T ; M = gamma @ wm (WMMA);
//           beta' = mean + eps2*svm - M@mean  (folds the xm mean-subtraction
//           out of the output GEMM: y = M@xs + beta').  One WG per batch.
// ---------------------------------------------------------------------------
__global__ __launch_bounds__(256) void k_gamma(const float* __restrict__ wm,
                                               const float* __restrict__ svw,
                                               const float* __restrict__ eps1,
                                               const float* __restrict__ meanv,
                                               const float* __restrict__ svm,
                                               const float* __restrict__ eps2,
                                               float* __restrict__ Mout,
                                               float* __restrict__ betaout) {
  __shared__ float sG[GN * GN];
  __shared__ float sW[GN * GN];
  __shared__ float sMo[GN * GN];
  const int b = blockIdx.x, tid = threadIdx.x;
  for (int e = tid; e < GN * GN; e += 256) {
    const int i = e >> 6, j = e & 63;
    const int src = (j >= i) ? e : (j * GN + i);   // symmetrize via triu
    sG[e] = wm[(size_t)b * GN * GN + src] + eps1[(size_t)b * GN * GN + src] * svw[src];
    sW[e] = wm[(size_t)b * GN * GN + e];
  }
  __syncthreads();
  mm64_wmma(sMo, sG, sW, tid);                     // M = gamma @ wm
  __syncthreads();
  for (int e = tid; e < GN * GN; e += 256)
    Mout[(size_t)b * GN * GN + e] = sMo[e];
  if (tid < GN) {
    float acc = meanv[b * GN + tid] + eps2[b * GN + tid] * svm[tid];
#pragma unroll 8
    for (int h = 0; h < GN; ++h)
      acc -= sMo[tid * GN + h] * meanv[b * GN + h]; // fold -M@mean into beta
    betaout[b * GN + tid] = acc;
  }
}

// ---------------------------------------------------------------------------
// Kernel 5: y = M @ xs + beta'  (mean already folded into beta'),
//           scattered back through idx.
//           Grid: (128 chunks of 128 cols, 32 batches); 256 threads.
// ---------------------------------------------------------------------------
__global__ __launch_bounds__(256) void k_out(const float* __restrict__ x,
                                             const int* __restrict__ idx,
                                             const float* __restrict__ Mmat,
                                             const float* __restrict__ betav,
                                             float* __restrict__ out) {
  __shared__ __align__(16) float sX[GN * 128];   // 32 KB : xs chunk [h][c]
  __shared__ __align__(16) float sM[GN * GN];    // 16 KB
  __shared__ float sBeta[GN];
  __shared__ int   sChan[GN];
  const int chunk = blockIdx.x;          // 0..127
  const int b     = blockIdx.y;
  const int cg    = chunk >> 5;
  const int sbase = (chunk & 31) * 128;
  const int tid   = threadIdx.x;

  // async-stage M (1024 x b128, 4/thread); fill chan/beta tables
  const float* Mg = Mmat + (size_t)b * GN * GN;
  const unsigned sMb = lds_off(sM);
  for (int e = tid; e < 1024; e += 256)
    async_load_b128(sMb + (unsigned)e * 16u, Mg + e * 4);
  if (tid < GN) {
    sBeta[tid] = betav[b * GN + tid];
    sChan[tid] = idx[4 * tid + cg];
  }
  __syncthreads();   // sChan visible before use below

  // async-stage xs chunk (2048 x b128, 8/thread), raw (no mean subtract)
  const unsigned sXb = lds_off(sX);
  for (int e = tid; e < 2048; e += 256) {
    const int h  = e >> 5;            // row
    const int c4 = e & 31;            // 16B chunk within 128-col row
    async_load_b128(sXb + (unsigned)e * 16u,
                    x + ((size_t)(b * CN + sChan[h])) * HWN + sbase + c4 * 4);
  }
  wait_async0();
  __syncthreads();

  const int lane = tid & 31;
  const int wave = tid >> 5;
  const int l15  = lane & 15;
  const int koff = (lane < 16) ? 0 : 2;

  // 4 row-tiles x 8 col-tiles = 32 tiles; 4 per wave
  for (int t = wave; t < 32; t += 8) {
    const int g0 = (t >> 3) << 4;
    const int n0 = (t & 7) << 4;
    v8f acc = {};
#pragma unroll 4
    for (int kk = 0; kk < GN; kk += 4) {
      v2f av, bv;
      av.x = sM[(g0 + l15) * GN + kk + koff];
      av.y = sM[(g0 + l15) * GN + kk + koff + 1];
      bv.x = sX[(kk + koff) * 128 + n0 + l15];
      bv.y = sX[(kk + koff + 1) * 128 + n0 + l15];
      acc = __builtin_amdgcn_wmma_f32_16x16x4_f32(false, av, false, bv,
                                                  (short)0, acc, false, false);
    }
    const int mb = g0 + ((lane < 16) ? 0 : 8);
    const int s  = sbase + n0 + l15;
#pragma unroll
    for (int r = 0; r < 8; ++r) {
      const int g = mb + r;
      out[((size_t)(b * CN + sChan[g])) * HWN + s] = acc[r] + sBeta[g];
    }
  }
}

// ---------------------------------------------------------------------------
extern "C" void kernel_launch(void* const* d_in, const int* in_sizes, int n_in,
                              void* d_out, int out_size, void* d_ws, size_t ws_size,
                              hipStream_t stream) {
  (void)in_sizes; (void)n_in; (void)out_size; (void)ws_size;
  const float* x    = (const float*)d_in[0];
  const int*   idx  = (const int*)d_in[1];
  const float* eps1 = (const float*)d_in[2];
  const float* eps2 = (const float*)d_in[3];
  float* out = (float*)d_out;

  float* ws     = (float*)d_ws;
  float* gram   = ws;                       // B*G*G = 131072
  float* rowsum = gram + BN * GN * GN;      // B*G   = 2048
  float* wmv    = rowsum + BN * GN;         // B*G*G
  float* meanv  = wmv + BN * GN * GN;       // B*G
  float* Mmat   = meanv + BN * GN;          // B*G*G
  float* betav  = Mmat + BN * GN * GN;      // B*G
  float* svw    = betav + BN * GN;          // G*G = 4096
  float* svm    = svw + GN * GN;            // G   = 64

  const int nz = BN * GN * GN + BN * GN;    // gram + rowsum
  k_zero<<<(nz + 255) / 256, 256, 0, stream>>>(gram, nz);
  k_gram<<<dim3(64, BN), 256, 0, stream>>>(x, idx, gram, rowsum);
  k_newton<<<BN, 256, 0, stream>>>(gram, rowsum, wmv, meanv);
  k_stats<<<(GN * GN + GN + 255) / 256, 256, 0, stream>>>(wmv, meanv, svw, svm);
  k_gamma<<<BN, 256, 0, stream>>>(wmv, svw, eps1, meanv, svm, eps2, Mmat, betav);
  k_out<<<dim3(128, BN), 256, 0, stream>>>(x, idx, Mmat, betav, out);
}